// L2LGD2_19705309954331
// MI455X (gfx1250) — compile-verified
//
#include <hip/hip_runtime.h>
#include <math.h>

typedef float v2f __attribute__((ext_vector_type(2)));
typedef float v8f __attribute__((ext_vector_type(8)));

#define HID 8
#define TILES_PER_WAVE 4
#define WAVES_PER_BLOCK 8
#define EXP_P_CONST 22026.465794806718f   // exp(float32(10))
#define LOG2E_F     1.4426950408889634f
#define N2LOG2E_F  -2.8853900817779268f   // -2*log2(e)
#define LN2_OVER_10 0.069314718055994531f

// ds_swizzle BITMASK_PERM: src_lane = lane ^ m  (offset[14:10]=xor, [4:0]=and=31)
#define SWZ_XOR(x, m) \
    __int_as_float(__builtin_amdgcn_ds_swizzle(__float_as_int(x), (0x1F | ((m) << 10))))

__device__ __forceinline__ float rcp_raw(float x)  { return __builtin_amdgcn_rcpf(x); }
__device__ __forceinline__ float exp2_raw(float x) { return __builtin_amdgcn_exp2f(x); }
__device__ __forceinline__ float log2_raw(float x) { return __builtin_amdgcn_logf(x); }
// sigmoid(x) = 1/(1+2^(-x*log2e)) : mul, exp2, add, rcp
__device__ __forceinline__ float fast_sigmoid(float x) {
    return rcp_raw(1.0f + exp2_raw(x * -LOG2E_F));
}
// tanh(x) = 2/(1+2^(-2x*log2e)) - 1 ; saturates correctly at +-1
__device__ __forceinline__ float fast_tanh(float x) {
    return fmaf(2.0f, rcp_raw(1.0f + exp2_raw(x * N2LOG2E_F)), -1.0f);
}

__device__ __forceinline__ v2f make_v2f(float x, float y) { v2f v; v.x = x; v.y = y; return v; }

__device__ __forceinline__ v8f splat8(float x) {
    v8f v;
#pragma unroll
    for (int i = 0; i < 8; ++i) v[i] = x;
    return v;
}

__device__ __forceinline__ v8f wmma4(v2f a, v2f b, v8f c) {
    // D(16x16 f32) = A(16x4 f32) * B(4x16 f32) + C
    return __builtin_amdgcn_wmma_f32_16x16x4_f32(false, a, false, b, (short)0, c, false, false);
}

// One wave handles TILES_PER_WAVE tiles of 16 rows.
// Chain 1 packs both sigmoid gates: B cols 0-7 = Wh_r, cols 8-15 = Wh_z (same A).
// Chain 2: t_n = h@Wh_n + bh_n ; u_n = g_in@Wi_n (cols 0-7).
__global__ __launch_bounds__(256) void gru_wmma_kernel(
    const float* __restrict__ h, const float* __restrict__ g,
    const float* __restrict__ Wi_r, const float* __restrict__ Wh_r, const float* __restrict__ bh_r,
    const float* __restrict__ Wi_z, const float* __restrict__ Wh_z, const float* __restrict__ bh_z,
    const float* __restrict__ Wi_n, const float* __restrict__ Wh_n, const float* __restrict__ bh_n,
    const float* __restrict__ Wm, const float* __restrict__ bm,
    float* __restrict__ h_new, float* __restrict__ out, int n)
{
    const int  lane   = threadIdx.x & 31;
    const int  wave   = threadIdx.x >> 5;
    const int  j      = lane & 15;              // D-layout column / A-layout row
    const int  hiHalf = lane >> 4;              // K pair (A/B) or M half (C/D)
    const int  kofs   = hiHalf * 2;
    const int  c      = j & 7;                  // hid column 0..7
    const bool colOK  = (j < 8);
    const bool wiOK   = colOK && (hiHalf == 0); // Wi has K rows 0,1 only

    // ---- per-wave constant operands ----
    const float* Wrz = colOK ? Wh_r : Wh_z;     // r|z column packing
    const float* Wiz = colOK ? Wi_r : Wi_z;
    const float* brz = colOK ? bh_r : bh_z;
    v2f B1_lo = make_v2f(Wrz[(kofs    ) * HID + c], Wrz[(kofs + 1) * HID + c]);
    v2f B1_hi = make_v2f(Wrz[(kofs + 4) * HID + c], Wrz[(kofs + 5) * HID + c]);
    v2f B1_g  = make_v2f(hiHalf ? 0.0f : Wiz[c], hiHalf ? 0.0f : Wiz[HID + c]);
    v2f B2_lo = make_v2f(colOK ? Wh_n[(kofs    ) * HID + c] : 0.0f,
                         colOK ? Wh_n[(kofs + 1) * HID + c] : 0.0f);
    v2f B2_hi = make_v2f(colOK ? Wh_n[(kofs + 4) * HID + c] : 0.0f,
                         colOK ? Wh_n[(kofs + 5) * HID + c] : 0.0f);
    v2f B2_g  = make_v2f(wiOK ? Wi_n[c] : 0.0f, wiOK ? Wi_n[HID + c] : 0.0f);
    v8f C1 = splat8(brz[c]);
    v8f C2 = splat8(colOK ? bh_n[c] : 0.0f);
    v8f C0 = splat8(0.0f);
    const float wm     = Wm[c];
    const float bm_l2  = bm[0] * LOG2E_F;               // folded into final exp2
    const int   bpAddr = (hiHalf * 8 + c) << 2;         // ds_bpermute byte address

    int rowBase = (blockIdx.x * WAVES_PER_BLOCK + wave) * TILES_PER_WAVE * 16;

    // incremental pointers (A base has kofs folded in -> immediate offsets 0/4/16/20)
    const float* hpA = h + (size_t)(rowBase + j) * HID + kofs;
    const float* gp  = g + rowBase + j;
    const float* hDp = h     + ((size_t)(rowBase + hiHalf * 8)) * HID + c;
    float*       hNp = h_new + ((size_t)(rowBase + hiHalf * 8)) * HID + c;
    float*       op  = out + rowBase + hiHalf * 8 + c;

#pragma unroll 1
    for (int t = 0; t < TILES_PER_WAVE; ++t) {
        if (rowBase + 16 > n) break;            // wave-uniform: EXEC all-ones for WMMA

        // ---- A operands: h rows (two K chunks), coalesced b64 loads ----
        v2f A_lo = make_v2f(hpA[0], hpA[1]);    // K = 0..3
        v2f A_hi = make_v2f(hpA[4], hpA[5]);    // K = 4..7

        // ---- g_in per row; third K-chunk (K=2,3 zero padded) ----
        float g_row = *gp;
        // |g|+1e-8 is never denormal -> raw v_log_f32 is safe; fold ln2/P into one mul
        float glog  = log2_raw(fabsf(g_row) + 1e-8f) * LN2_OVER_10;
        float gi0   = fmaxf(glog, -1.0f);       // == big ? glog : -1
        // when glog >= -1, g != 0, so sign(g) == copysign(1,g)
        float gi1   = (glog >= -1.0f) ? copysignf(1.0f, g_row) : EXP_P_CONST * g_row;
        v2f A_g = make_v2f(hiHalf ? 0.0f : gi0, hiHalf ? 0.0f : gi1);

        // ---- WMMA chains (6 per tile) ----
        v8f pre1 = wmma4(A_g, B1_g, wmma4(A_hi, B1_hi, wmma4(A_lo, B1_lo, C1))); // r|z
        v8f t_n  =                  wmma4(A_hi, B2_hi, wmma4(A_lo, B2_lo, C2));  // h@Wh_n+bh_n
        v8f u_n  = wmma4(A_g, B2_g, C0);                                         // g_in@Wi_n

        // ---- pointwise gates + h_new (full EXEC: shuffles need both halves) ----
        float hn[8];
#pragma unroll
        for (int v = 0; v < 8; ++v) {
            float sg = fast_sigmoid(pre1[v]);   // r at j<8, z at j>=8 (all lanes useful)
            float zz = SWZ_XOR(sg, 8);          // z for my column at j<8
            float nn = fast_tanh(fmaf(sg, t_n[v], u_n[v]));
            float hv = hDp[v * HID];
            hn[v] = fmaf(zz, hv - nn, nn);      // (1-z)*n + z*h
        }
        // g for row hiHalf*8+c (source lanes 0..15: gather while EXEC is full)
        float gsel = __int_as_float(__builtin_amdgcn_ds_bpermute(bpAddr, __float_as_int(g_row)));

        // ---- single predicated region: stores + transpose-reduce + out ----
        if (colOK) {
            float acc[8];
#pragma unroll
            for (int v = 0; v < 8; ++v) {
                __builtin_nontemporal_store(hn[v], &hNp[v * HID]);
                acc[v] = hn[v] * wm;
            }
            // transpose-reduce over 8 column lanes (xor partners stay in active set):
            // lane j ends with sum for row hiHalf*8 + j
            float b4[4];
            {
                const bool bit = (j & 1) != 0;
#pragma unroll
                for (int i = 0; i < 4; ++i) {
                    float give = bit ? acc[2 * i]     : acc[2 * i + 1];
                    float keep = bit ? acc[2 * i + 1] : acc[2 * i];
                    b4[i] = keep + SWZ_XOR(give, 1);
                }
            }
            float c2[2];
            {
                const bool bit = (j & 2) != 0;
#pragma unroll
                for (int i = 0; i < 2; ++i) {
                    float give = bit ? b4[2 * i]     : b4[2 * i + 1];
                    float keep = bit ? b4[2 * i + 1] : b4[2 * i];
                    c2[i] = keep + SWZ_XOR(give, 2);
                }
            }
            float total;
            {
                const bool bit = (j & 4) != 0;
                float give = bit ? c2[0] : c2[1];
                float keep = bit ? c2[1] : c2[0];
                total = keep + SWZ_XOR(give, 4);
            }
            // -exp(total+bm)*g  ==  -2^(total*log2e + bm*log2e) * g
            float o = -exp2_raw(fmaf(total, LOG2E_F, bm_l2)) * gsel;
            __builtin_nontemporal_store(o, op);
        }

        rowBase += 16;
        hpA += 16 * HID; hDp += 16 * HID; hNp += 16 * HID; gp += 16; op += 16;
    }
}

// Scalar tail for n % 16 rows (not launched when N is a multiple of 16).
__global__ void gru_tail_kernel(
    const float* __restrict__ h, const float* __restrict__ g,
    const float* __restrict__ Wi_r, const float* __restrict__ Wh_r, const float* __restrict__ bh_r,
    const float* __restrict__ Wi_z, const float* __restrict__ Wh_z, const float* __restrict__ bh_z,
    const float* __restrict__ Wi_n, const float* __restrict__ Wh_n, const float* __restrict__ bh_n,
    const float* __restrict__ Wm, const float* __restrict__ bm,
    float* __restrict__ h_new, float* __restrict__ out, int start, int n)
{
    int row = start + blockIdx.x * blockDim.x + threadIdx.x;
    if (row >= n) return;
    float hv[HID];
#pragma unroll
    for (int i = 0; i < HID; ++i) hv[i] = h[(size_t)row * HID + i];
    float g_row = g[row];
    float glog  = log2_raw(fabsf(g_row) + 1e-8f) * LN2_OVER_10;
    float gi0   = fmaxf(glog, -1.0f);
    float gi1   = (glog >= -1.0f) ? copysignf(1.0f, g_row) : EXP_P_CONST * g_row;

    float r8[HID], z8[HID];
#pragma unroll
    for (int cc = 0; cc < HID; ++cc) {
        float ar = gi0 * Wi_r[cc] + gi1 * Wi_r[HID + cc] + bh_r[cc];
        float az = gi0 * Wi_z[cc] + gi1 * Wi_z[HID + cc] + bh_z[cc];
#pragma unroll
        for (int k = 0; k < HID; ++k) {
            ar += hv[k] * Wh_r[k * HID + cc];
            az += hv[k] * Wh_z[k * HID + cc];
        }
        r8[cc] = fast_sigmoid(ar);
        z8[cc] = fast_sigmoid(az);
    }
    float outs = bm[0];
#pragma unroll
    for (int cc = 0; cc < HID; ++cc) {
        float an = gi0 * Wi_n[cc] + gi1 * Wi_n[HID + cc];
        float tn = bh_n[cc];
#pragma unroll
        for (int k = 0; k < HID; ++k) tn += hv[k] * Wh_n[k * HID + cc];
        float nn = fast_tanh(an + r8[cc] * tn);
        float hn = fmaf(z8[cc], hv[cc] - nn, nn);
        h_new[(size_t)row * HID + cc] = hn;
        outs += hn * Wm[cc];
    }
    out[row] = -exp2_raw(fmaf(outs, LOG2E_F, 0.0f)) * g_row;
}

extern "C" void kernel_launch(void* const* d_in, const int* in_sizes, int n_in,
                              void* d_out, int out_size, void* d_ws, size_t ws_size,
                              hipStream_t stream) {
    const float* h    = (const float*)d_in[0];
    const float* g    = (const float*)d_in[1];
    const float* Wi_r = (const float*)d_in[2];
    const float* Wh_r = (const float*)d_in[3];
    const float* bh_r = (const float*)d_in[4];
    const float* Wi_z = (const float*)d_in[5];
    const float* Wh_z = (const float*)d_in[6];
    const float* bh_z = (const float*)d_in[7];
    const float* Wi_n = (const float*)d_in[8];
    const float* Wh_n = (const float*)d_in[9];
    const float* bh_n = (const float*)d_in[10];
    const float* Wm   = (const float*)d_in[11];
    const float* bm   = (const float*)d_in[12];
    const int n = in_sizes[1];                       // g has N elements

    float* h_new = (float*)d_out;                    // [N,8]
    float* out   = (float*)d_out + (size_t)n * HID;  // [N]

    const int nTiles        = n / 16;
    const int tilesPerBlock = WAVES_PER_BLOCK * TILES_PER_WAVE;  // 32
    const int blocks        = (nTiles + tilesPerBlock - 1) / tilesPerBlock;
    if (blocks > 0) {
        gru_wmma_kernel<<<blocks, 256, 0, stream>>>(
            h, g, Wi_r, Wh_r, bh_r, Wi_z, Wh_z, bh_z, Wi_n, Wh_n, bh_n,
            Wm, bm, h_new, out, n);
    }
    const int done = nTiles * 16;
    const int rem  = n - done;
    if (rem > 0) {
        gru_tail_kernel<<<(rem + 255) / 256, 256, 0, stream>>>(
            h, g, Wi_r, Wh_r, bh_r, Wi_z, Wh_z, bh_z, Wi_n, Wh_n, bh_n,
            Wm, bm, h_new, out, done, n);
    }
}